// GCN_75496935129706
// MI455X (gfx1250) — compile-verified
//
#include <hip/hip_runtime.h>
#include <hip/hip_bf16.h>
#include <math.h>

typedef __attribute__((ext_vector_type(2))) float v2f;
typedef __attribute__((ext_vector_type(8))) float v8f;

#define N_NODES 100000
#define N_EDGES 1600000
#define HIDDEN  64
#define N_GRAPHS 64
#define IN_FEAT 5

// ---------------- utility ----------------
__global__ __launch_bounds__(256) void zero_f(float* __restrict__ p, int n) {
  int i = blockIdx.x * 256 + threadIdx.x;
  if (i < n) p[i] = 0.0f;
}

// degree over target index (col); one thread per edge
__global__ __launch_bounds__(256) void deg_kernel(const long long* __restrict__ col,
                                                  float* __restrict__ cnt, int E) {
  int e = blockIdx.x * 256 + threadIdx.x;
  if (e >= E) return;
  long long c = col[e];
  __hip_atomic_fetch_add(cnt + c, 1.0f, __ATOMIC_RELAXED, __HIP_MEMORY_SCOPE_AGENT);
}

// dis[i] = rsqrt(deg[i] + 1)   (self-loop included; deg always > 0)
__global__ __launch_bounds__(256) void dis_kernel(const float* __restrict__ cnt,
                                                  float* __restrict__ dis, int n) {
  int i = blockIdx.x * 256 + threadIdx.x;
  if (i < n) dis[i] = rsqrtf(cnt[i] + 1.0f);
}

// ---------------- layer 1 GEMM: [N,5] @ [5,64] (K too small for WMMA) -------
__global__ __launch_bounds__(256) void gemm5(const float* __restrict__ x,
                                             const float* __restrict__ W1,
                                             float* __restrict__ out, int total) {
  int gid = blockIdx.x * 256 + threadIdx.x;
  if (gid >= total) return;
  int i = gid >> 6, j = gid & 63;
  const float* xr = x + (size_t)i * IN_FEAT;
  float s = 0.0f;
#pragma unroll
  for (int k = 0; k < IN_FEAT; ++k) s += xr[k] * W1[k * HIDDEN + j];
  out[gid] = s;
}

// ---------------- hidden GEMM: [N,64] @ [64,64] via V_WMMA_F32_16X16X4_F32 --
// block = 256 threads = 8 waves; covers 32 rows x 64 cols
// wave w: rowTile = w>>2 (16 rows), colTile = w&3 (16 cols)
__global__ __launch_bounds__(256) void gemm64_wmma(const float* __restrict__ H,
                                                   const float* __restrict__ W,
                                                   float* __restrict__ out) {
  __shared__ float lw[HIDDEN * HIDDEN];
  for (int i = threadIdx.x; i < HIDDEN * HIDDEN; i += 256) lw[i] = W[i];
  __syncthreads();

  const int wave = threadIdx.x >> 5;
  const int lane = threadIdx.x & 31;
  const int rowTile = wave >> 2;   // 0..1
  const int colTile = wave & 3;    // 0..3
  const int row0 = blockIdx.x * 32 + rowTile * 16;
  const int m = lane & 15;         // row within A tile / col within B tile
  const int half = lane >> 4;      // selects K pair

  v8f acc = {};
  const float* hrow = H + (size_t)(row0 + m) * HIDDEN;

#pragma unroll
  for (int kk = 0; kk < 16; ++kk) {
    const int k0 = kk * 4;
    // A (16x4): lanes 0-15 hold K=k0,k0+1; lanes 16-31 hold K=k0+2,k0+3
    v2f a;
    a.x = hrow[k0 + 2 * half];
    a.y = hrow[k0 + 2 * half + 1];
    // B (4x16): lane n = m; lanes 0-15 hold K=k0,k0+1; 16-31 hold K=k0+2,k0+3
    v2f b;
    b.x = lw[(k0 + 2 * half) * HIDDEN + colTile * 16 + m];
    b.y = lw[(k0 + 2 * half + 1) * HIDDEN + colTile * 16 + m];
    acc = __builtin_amdgcn_wmma_f32_16x16x4_f32(
        /*neg_a=*/false, a, /*neg_b=*/false, b,
        /*c_mod=*/(short)0, acc, /*reuse_a=*/false, /*reuse_b=*/false);
  }

  // D layout: VGPR r -> row (r + 8*half), col = m
  float* orow = out + (size_t)row0 * HIDDEN + colTile * 16 + m;
#pragma unroll
  for (int r = 0; r < 8; ++r)
    orow[(size_t)(r + 8 * half) * HIDDEN] = acc[r];
}

// AGG[i][j] = HW[i][j]*dis[i]^2 + b[j]   (self-loop msg + bias, doubles as init)
__global__ __launch_bounds__(256) void init_agg(const float* __restrict__ HW,
                                                const float* __restrict__ dis,
                                                const float* __restrict__ b,
                                                float* __restrict__ AGG, int total) {
  int gid = blockIdx.x * 256 + threadIdx.x;
  if (gid >= total) return;
  int i = gid >> 6, j = gid & 63;
  float d = dis[i];
  AGG[gid] = HW[gid] * d * d + b[j];
}

// one wave (32 lanes) per edge: gather 64 floats of HW[row], scaled atomic-add
// into AGG[col]
__global__ __launch_bounds__(256) void scatter_edges(const long long* __restrict__ ei,
                                                     const float* __restrict__ dis,
                                                     const float* __restrict__ HW,
                                                     float* __restrict__ AGG, int E) {
  int e = (blockIdx.x * 256 + threadIdx.x) >> 5;
  int lane = threadIdx.x & 31;
  if (e >= E) return;
  long long r = ei[e];               // source
  long long c = ei[(size_t)E + e];   // target
  float nrm = dis[r] * dis[c];
  const float2* src = (const float2*)(HW + (size_t)r * HIDDEN);
  float2 v = src[lane];
  float* dst = AGG + (size_t)c * HIDDEN + 2 * lane;
  __hip_atomic_fetch_add(dst,     v.x * nrm, __ATOMIC_RELAXED, __HIP_MEMORY_SCOPE_AGENT);
  __hip_atomic_fetch_add(dst + 1, v.y * nrm, __ATOMIC_RELAXED, __HIP_MEMORY_SCOPE_AGENT);
}

__global__ __launch_bounds__(256) void tanh_inplace(float* __restrict__ p, int n) {
  int i = blockIdx.x * 256 + threadIdx.x;
  if (i < n) p[i] = tanhf(p[i]);
}

// one wave per node: accumulate into per-graph sums + counts
__global__ __launch_bounds__(256) void pool_kernel(const float* __restrict__ H,
                                                   const long long* __restrict__ batch,
                                                   float* __restrict__ sums,
                                                   float* __restrict__ counts, int N) {
  int i = (blockIdx.x * 256 + threadIdx.x) >> 5;
  int lane = threadIdx.x & 31;
  if (i >= N) return;
  long long g = batch[i];
  const float2* src = (const float2*)(H + (size_t)i * HIDDEN);
  float2 v = src[lane];
  float* dst = sums + g * HIDDEN + 2 * lane;
  __hip_atomic_fetch_add(dst,     v.x, __ATOMIC_RELAXED, __HIP_MEMORY_SCOPE_AGENT);
  __hip_atomic_fetch_add(dst + 1, v.y, __ATOMIC_RELAXED, __HIP_MEMORY_SCOPE_AGENT);
  if (lane == 0)
    __hip_atomic_fetch_add(counts + g, 1.0f, __ATOMIC_RELAXED, __HIP_MEMORY_SCOPE_AGENT);
}

// single block: pooled = sums/max(counts,1); tanh(pooled@l1W+l1b); tanh(@l2W+l2b)
__global__ __launch_bounds__(256) void head_kernel(const float* __restrict__ sums,
                                                   const float* __restrict__ counts,
                                                   const float* __restrict__ l1W,
                                                   const float* __restrict__ l1b,
                                                   const float* __restrict__ l2W,
                                                   const float* __restrict__ l2b,
                                                   float* __restrict__ out) {
  __shared__ float pooled[N_GRAPHS * HIDDEN];
  __shared__ float h1[N_GRAPHS * HIDDEN];
  for (int i = threadIdx.x; i < N_GRAPHS * HIDDEN; i += 256) {
    int g = i >> 6;
    pooled[i] = sums[i] / fmaxf(counts[g], 1.0f);
  }
  __syncthreads();
  for (int i = threadIdx.x; i < N_GRAPHS * HIDDEN; i += 256) {
    int g = i >> 6, j = i & 63;
    float s = l1b[j];
    for (int k = 0; k < HIDDEN; ++k) s += pooled[g * HIDDEN + k] * l1W[k * HIDDEN + j];
    h1[i] = tanhf(s);
  }
  __syncthreads();
  if (threadIdx.x < N_GRAPHS) {
    int g = threadIdx.x;
    float s = l2b[0];
    for (int j = 0; j < HIDDEN; ++j) s += h1[g * HIDDEN + j] * l2W[j];
    out[g] = tanhf(s);
  }
}

// ---------------------------------------------------------------------------
extern "C" void kernel_launch(void* const* d_in, const int* in_sizes, int n_in,
                              void* d_out, int out_size, void* d_ws, size_t ws_size,
                              hipStream_t stream) {
  const float*     x      = (const float*)d_in[0];
  const long long* ei     = (const long long*)d_in[1];  // [2, E] int64
  const long long* batch  = (const long long*)d_in[2];  // [N] int64
  const float*     W1     = (const float*)d_in[3];
  const float*     b1     = (const float*)d_in[4];
  const float*     Wh     = (const float*)d_in[5];      // [8,64,64]
  const float*     bh     = (const float*)d_in[6];      // [8,64]
  const float*     lin1_W = (const float*)d_in[7];
  const float*     lin1_b = (const float*)d_in[8];
  const float*     lin2_W = (const float*)d_in[9];
  const float*     lin2_b = (const float*)d_in[10];
  float* out = (float*)d_out;

  const int N = N_NODES, E = N_EDGES;
  const int NH = N * HIDDEN;  // 6,400,000

  char* ws = (char*)d_ws;
  const size_t NB = (size_t)NH * sizeof(float);  // 25.6 MB
  float* bufA   = (float*)ws;              // H ping
  float* bufB   = (float*)(ws + NB);       // HW scratch
  float* bufC   = (float*)(ws + 2 * NB);   // H pong
  float* cnt    = (float*)(ws + 3 * NB);   // [N] degree counts
  float* dis    = cnt + N;                 // [N]
  float* sums   = dis + N;                 // [64*64]
  float* counts = sums + N_GRAPHS * HIDDEN;// [64]

  const int gN   = (N + 255) / 256;        // 391
  const int gE   = (E + 255) / 256;        // 6250
  const int gNH  = (NH + 255) / 256;       // 25000
  const int gEw  = (E + 7) / 8;            // 200000 (wave per edge)
  const int gNw  = (N + 7) / 8;            // 12500  (wave per node)
  const int gMM  = N / 32;                 // 3125   (exact: 100000 % 32 == 0)

  // degrees / normalization
  zero_f<<<gN, 256, 0, stream>>>(cnt, N);
  zero_f<<<17, 256, 0, stream>>>(sums, N_GRAPHS * HIDDEN + N_GRAPHS);
  deg_kernel<<<gE, 256, 0, stream>>>(ei + E, cnt, E);   // deg over targets
  dis_kernel<<<gN, 256, 0, stream>>>(cnt, dis, N);

  // conv 1: x @ W1
  gemm5<<<gNH / 256 + 1, 256, 0, stream>>>(x, W1, bufB, NH);
  init_agg<<<gNH, 256, 0, stream>>>(bufB, dis, b1, bufC, NH);
  scatter_edges<<<gEw, 256, 0, stream>>>(ei, dis, bufB, bufC, E);
  tanh_inplace<<<gNH, 256, 0, stream>>>(bufC, NH);

  float* Hcur = bufC;
  float* Hnxt = bufA;
  for (int l = 0; l < 8; ++l) {
    gemm64_wmma<<<gMM, 256, 0, stream>>>(Hcur, Wh + (size_t)l * HIDDEN * HIDDEN, bufB);
    init_agg<<<gNH, 256, 0, stream>>>(bufB, dis, bh + (size_t)l * HIDDEN, Hnxt, NH);
    scatter_edges<<<gEw, 256, 0, stream>>>(ei, dis, bufB, Hnxt, E);
    tanh_inplace<<<gNH, 256, 0, stream>>>(Hnxt, NH);
    float* t = Hcur; Hcur = Hnxt; Hnxt = t;
  }

  // global mean pool + MLP head
  pool_kernel<<<gNw, 256, 0, stream>>>(Hcur, batch, sums, counts, N);
  head_kernel<<<1, 256, 0, stream>>>(sums, counts, lin1_W, lin1_b, lin2_W, lin2_b, out);
}